// Attention_47837345743160
// MI455X (gfx1250) — compile-verified
//
#include <hip/hip_runtime.h>

#define N_MEL 80
#define N_CTX 640
#define N_ATT 128
#define TQ    512
#define TK    256
#define BATCH 8

typedef float v2f __attribute__((ext_vector_type(2)));
typedef float v8f __attribute__((ext_vector_type(8)));

#define KST 132   // padded LDS stride (floats) for 128-float rows; 132%64=4 ->
                  // the 16 k-lanes' float4 reads tile all 64 banks (conflict-free)
#define PST 260   // padded LDS stride for 256-float rows; 260%64=4 -> conflict-free

// exp(2x) = exp2(x * 2*log2(e))
__device__ __forceinline__ float tanh_fast(float x) {
  x = fminf(fmaxf(x, -15.f), 15.f);
  float t = __builtin_amdgcn_exp2f(x * 2.8853900817779268f);   // v_exp_f32
  return 1.f - 2.f * __builtin_amdgcn_rcpf(t + 1.f);           // v_rcp_f32
}

// ---------------------------------------------------------------------------
// Kernel 1: fused projections  qp = queries @ Wq^T, kp = keys @ Wk^T,
// vp = values @ Wv^T, laid out batch-major: qp[b][t][c], kp[b][k][c], vp[b][k][c]
// ---------------------------------------------------------------------------
__global__ __launch_bounds__(128) void proj_kernel(
    const float* __restrict__ queries, const float* __restrict__ keys,
    const float* __restrict__ values,
    const float* __restrict__ Wq, const float* __restrict__ Wk,
    const float* __restrict__ Wv,
    float* __restrict__ qp, float* __restrict__ kp, float* __restrict__ vp)
{
  __shared__ float srow[N_CTX];
  const int r = blockIdx.x, tid = threadIdx.x;
  const float* in; const float* W; float* outp; int IN;
  if (r < BATCH * TQ) {                         // q rows: r = b*TQ + t
    int b = r / TQ, t = r % TQ;
    in = queries + (size_t)(t * BATCH + b) * N_MEL;
    W = Wq; IN = N_MEL; outp = qp + (size_t)r * N_ATT;
  } else if (r < BATCH * TQ + BATCH * TK) {     // k rows
    int r2 = r - BATCH * TQ;
    int b = r2 / TK, t = r2 % TK;
    in = keys + (size_t)(t * BATCH + b) * N_CTX;
    W = Wk; IN = N_CTX; outp = kp + (size_t)r2 * N_ATT;
  } else {                                      // v rows
    int r2 = r - BATCH * TQ - BATCH * TK;
    int b = r2 / TK, t = r2 % TK;
    in = values + (size_t)(t * BATCH + b) * N_CTX;
    W = Wv; IN = N_CTX; outp = vp + (size_t)r2 * N_ATT;
  }
  for (int i = tid; i < IN; i += 128) srow[i] = in[i];
  __syncthreads();
  const float* wrow = W + (size_t)tid * IN;     // out[c] = sum_i x[i] * W[c][i]
  float acc = 0.f;
  #pragma unroll 4
  for (int i = 0; i < IN; i += 4) {
    float4 w4 = *(const float4*)(wrow + i);
    acc = fmaf(srow[i + 0], w4.x, acc);
    acc = fmaf(srow[i + 1], w4.y, acc);
    acc = fmaf(srow[i + 2], w4.z, acc);
    acc = fmaf(srow[i + 3], w4.w, acc);
  }
  outp[tid] = acc;
}

// ---------------------------------------------------------------------------
// Kernel 2: per (b, 16-row q tile): energies + softmax + attn@V via f32 WMMA
// ---------------------------------------------------------------------------
__global__ __launch_bounds__(256) void attn_kernel(
    const float* __restrict__ qp, const float* __restrict__ kp,
    const float* __restrict__ vp, const float* __restrict__ Wvec,
    float* __restrict__ out, float* __restrict__ attn)
{
  extern __shared__ float smem[];
  float* ks    = smem;                 // TK * KST
  float* qs    = ks + TK * KST;        // 16 * KST
  float* wv    = qs + 16 * KST;        // N_ATT
  float* probs = wv + N_ATT;           // 16 * PST (16B-aligned base)

  const int tid = threadIdx.x;
  const int b   = blockIdx.x >> 5;     // TQ/16 = 32 tiles per batch
  const int qt  = blockIdx.x & 31;

  const float* kp_b = kp + (size_t)b * TK * N_ATT;
  const float* vp_b = vp + (size_t)b * TK * N_ATT;
  const float* qp_t = qp + ((size_t)b * TQ + qt * 16) * N_ATT;

  // Stage K-projection + q-tile + Wvec into LDS with float4 moves
  {
    const float4* src = (const float4*)kp_b;             // 256*32 float4
    for (int i4 = tid; i4 < TK * (N_ATT / 4); i4 += 256) {
      int r = i4 >> 5, c4 = i4 & 31;
      *(float4*)(ks + r * KST + c4 * 4) = src[i4];
    }
    const float4* srcq = (const float4*)qp_t;            // 16*32 float4
    for (int i4 = tid; i4 < 16 * (N_ATT / 4); i4 += 256) {
      int r = i4 >> 5, c4 = i4 & 31;
      *(float4*)(qs + r * KST + c4 * 4) = srcq[i4];
    }
    if (tid < N_ATT) wv[tid] = Wvec[tid];
  }
  __syncthreads();

  // ---- Energy: e[qi][k] = sum_c Wvec[c] * tanh(q[qi][c] + k[k][c]) ----
  const int kj = tid & 15, qi = tid >> 4;
  const float4* q4 = (const float4*)(qs + qi * KST);
  const float4* w4 = (const float4*)wv;
  float lmax = -3.4e38f;
  for (int s = 0; s < 16; ++s) {
    const int k = s * 16 + kj;
    const float4* k4 = (const float4*)(ks + k * KST);
    float acc = 0.f;
    #pragma unroll 4
    for (int c4 = 0; c4 < N_ATT / 4; ++c4) {
      float4 qv = q4[c4], kv = k4[c4], wc = w4[c4];
      acc = fmaf(wc.x, tanh_fast(qv.x + kv.x), acc);
      acc = fmaf(wc.y, tanh_fast(qv.y + kv.y), acc);
      acc = fmaf(wc.z, tanh_fast(qv.z + kv.z), acc);
      acc = fmaf(wc.w, tanh_fast(qv.w + kv.w), acc);
    }
    probs[qi * PST + k] = acc;
    lmax = fmaxf(lmax, acc);
  }

  // ---- Softmax over Tk: each 16-wide q-row lives in one wave32 half ->
  //      reduce max/sum with shfl_xor (no LDS, no barriers) ----
  #pragma unroll
  for (int off = 1; off < 16; off <<= 1)
    lmax = fmaxf(lmax, __shfl_xor(lmax, off));
  float lsum = 0.f;
  for (int s = 0; s < 16; ++s) {
    const int k = s * 16 + kj;
    float p = __builtin_amdgcn_exp2f((probs[qi * PST + k] - lmax) * 1.4426950408889634f);
    probs[qi * PST + k] = p;
    lsum += p;
  }
  #pragma unroll
  for (int off = 1; off < 16; off <<= 1)
    lsum += __shfl_xor(lsum, off);
  const float inv = 1.0f / lsum;
  float* attn_row = attn + ((size_t)b * TQ + qt * 16 + qi) * TK;
  for (int s = 0; s < 16; ++s) {
    const int k = s * 16 + kj;
    float val = probs[qi * PST + k] * inv;
    probs[qi * PST + k] = val;
    attn_row[k] = val;
  }
  __syncthreads();

  // ---- attn(16xTK) @ V(TKxN_ATT) via V_WMMA_F32_16X16X4_F32 ----
  // Wave w owns output columns [16w, 16w+16). A: lane L holds A[L%16][2*(L/16)+{0,1}].
  const int lane = tid & 31;
  const int n0   = (tid >> 5) * 16;
  const int mn   = lane & 15;       // A row m == B col n for this lane
  const int half = lane >> 4;       // selects K sub-pair {0,1} vs {2,3}
  v8f acc = {};
  for (int k0 = 0; k0 < TK; k0 += 4) {
    const int ka = k0 + 2 * half;
    v2f a, bv;
    a.x  = probs[mn * PST + ka];
    a.y  = probs[mn * PST + ka + 1];
    bv.x = vp_b[(size_t)ka * N_ATT + n0 + mn];
    bv.y = vp_b[(size_t)(ka + 1) * N_ATT + n0 + mn];
    acc = __builtin_amdgcn_wmma_f32_16x16x4_f32(
        false, a, false, bv, (short)0, acc, false, false);
  }
  // D layout: VGPR j -> M=j (lanes 0-15) / M=j+8 (lanes 16-31), N=lane%16.
  // Store directly transposed: out[b][c][t] (8 consecutive t per lane -> b128)
  float* out_b = out + (size_t)b * N_ATT * TQ;
  #pragma unroll
  for (int j = 0; j < 8; ++j) {
    const int mrow = j + half * 8;
    out_b[(size_t)(n0 + mn) * TQ + qt * 16 + mrow] = acc[j];
  }
}

// ---------------------------------------------------------------------------
extern "C" void kernel_launch(void* const* d_in, const int* in_sizes, int n_in,
                              void* d_out, int out_size, void* d_ws, size_t ws_size,
                              hipStream_t stream) {
  const float* queries = (const float*)d_in[0];
  const float* keys    = (const float*)d_in[1];
  const float* values  = (const float*)d_in[2];
  const float* Wq      = (const float*)d_in[3];
  const float* Wk      = (const float*)d_in[4];
  const float* Wv      = (const float*)d_in[5];
  const float* Wvec    = (const float*)d_in[6];

  float* qp = (float*)d_ws;                          // 8*512*128 = 2 MB
  float* kp = qp + (size_t)BATCH * TQ * N_ATT;       // 8*256*128 = 1 MB
  float* vp = kp + (size_t)BATCH * TK * N_ATT;       // 8*256*128 = 1 MB

  float* out_t  = (float*)d_out;                     // (B, C, Tq)
  float* attn_t = out_t + (size_t)BATCH * N_ATT * TQ;// (B, Tq, Tk)

  proj_kernel<<<BATCH * TQ + 2 * BATCH * TK, 128, 0, stream>>>(
      queries, keys, values, Wq, Wk, Wv, qp, kp, vp);

  const size_t lds_bytes =
      (size_t)(TK * KST + 16 * KST + N_ATT + 16 * PST) * sizeof(float);
  attn_kernel<<<BATCH * (TQ / 16), 256, lds_bytes, stream>>>(
      qp, kp, vp, Wvec, out_t, attn_t);
}